// GCNMultiplex_73813307949744
// MI455X (gfx1250) — compile-verified
//
#include <hip/hip_runtime.h>

#define NN   50000
#define LL   2
#define FIN  128
#define FOUT 32
#define ELEN 800000
#define NEG  0.2f

typedef float v2f __attribute__((ext_vector_type(2)));
typedef float v8f __attribute__((ext_vector_type(8)));

// ---------------------------------------------------------------------------
// Init: degree counts start at 2.0 (self-loop + cross-layer edge contribute
// exactly +1 each to every node's src count and trg count); zero agg buffer.
// deg buffer = 4*NN contiguous floats (deg_src | deg_trg), agg = 2*NN*FOUT.
// ---------------------------------------------------------------------------
__global__ __launch_bounds__(256) void k_init(float* __restrict__ deg,
                                              float* __restrict__ agg) {
    int i = blockIdx.x * 256 + threadIdx.x;
    if (i < 4 * NN) deg[i] = 2.0f;
    if (i < 2 * NN * FOUT) agg[i] = 0.0f;
}

// ---------------------------------------------------------------------------
// Degree counting over the 2*ELEN explicit edges (layer-1 edges offset by NN).
// ---------------------------------------------------------------------------
__global__ __launch_bounds__(256) void k_deg(const int* __restrict__ e0,
                                             const int* __restrict__ e1,
                                             float* __restrict__ dsrc,
                                             float* __restrict__ dtrg) {
    int t = blockIdx.x * 256 + threadIdx.x;
    if (t >= 2 * ELEN) return;
    int s, g;
    if (t < ELEN) {
        s = e0[t];
        g = e0[ELEN + t];
    } else {
        int u = t - ELEN;
        s = e1[u] + NN;
        g = e1[ELEN + u] + NN;
    }
    atomicAdd(&dsrc[s], 1.0f);
    atomicAdd(&dtrg[g], 1.0f);
}

// ---------------------------------------------------------------------------
// Projection GEMM, full f32 WMMA: P[l*NN+n, f] = rsqrt(deg_trg) * (x @ Wp^T)
// One wave per 16x16 output tile. K-loop: 128 / 4 = 32 WMMA steps.
// A layout (16x4 f32): lane L holds (M = L%16, K = 2*(L/16)+j) in comp j.
// B layout (4x16 f32): lane L holds (K = 2*(L/16)+j, Ncol = L%16) in comp j.
// C/D layout: lane L, comp v -> (M = v + 8*(L/16), Ncol = L%16).
// ---------------------------------------------------------------------------
__global__ __launch_bounds__(256) void k_proj(const float* __restrict__ x,
                                              const float* __restrict__ Wp,
                                              const float* __restrict__ dtrg,
                                              float* __restrict__ P) {
    int wid = blockIdx.x * 8 + (threadIdx.x >> 5);
    if (wid >= (NN / 16) * 4) return;          // 3125 row-tiles * 4 col-tiles
    int tm   = wid >> 2;
    int tn   = wid & 3;
    int lane = threadIdx.x & 31;
    int half = lane >> 4;
    int lid  = lane & 15;

    const float* xr = x  + (tm * 16 + lid) * FIN;  // A row (node)
    const float* wr = Wp + (tn * 16 + lid) * FIN;  // B col (out channel o)

    v8f acc = {};
    for (int k0 = 0; k0 < FIN; k0 += 4) {
        int ka = k0 + half * 2;
        v2f a; a.x = xr[ka]; a.y = xr[ka + 1];
        v2f b; b.x = wr[ka]; b.y = wr[ka + 1];
        acc = __builtin_amdgcn_wmma_f32_16x16x4_f32(false, a, false, b,
                                                    (short)0, acc, false, false);
    }

    int o = tn * 16 + lid;        // global out channel in [0,64)
    int l = o >> 5;               // layer
    int f = o & 31;               // feature within layer
#pragma unroll
    for (int v = 0; v < 8; ++v) {
        int node = tm * 16 + v + 8 * half;
        int vr   = l * NN + node;             // row of P (L*N layout)
        P[vr * FOUT + f] = acc[v] * rsqrtf(dtrg[vr]);
    }
}

// ---------------------------------------------------------------------------
// Edge aggregation: agg[trg] += P[src], one thread per (edge, feature).
// ---------------------------------------------------------------------------
__global__ __launch_bounds__(256) void k_edge(const int* __restrict__ e0,
                                              const int* __restrict__ e1,
                                              const float* __restrict__ P,
                                              float* __restrict__ agg) {
    long long t = (long long)blockIdx.x * 256 + threadIdx.x;
    if (t >= (long long)2 * ELEN * FOUT) return;
    int f = (int)(t & 31);
    int e = (int)(t >> 5);
    int s, g;
    if (e < ELEN) {
        s = e0[e];
        g = e0[ELEN + e];
    } else {
        int u = e - ELEN;
        s = e1[u] + NN;
        g = e1[ELEN + u] + NN;
    }
    atomicAdd(&agg[g * FOUT + f], P[s * FOUT + f]);
}

// ---------------------------------------------------------------------------
// Merge GEMM, full f32 WMMA, with fused epilogue-of-aggregation in the A load:
//   h(n,k) = leaky( (agg[vr] + P[vr] (self-loop) + P[partner] (cross-layer))
//                    * rsqrt(deg_src[vr]) + bias[k] )
//   out[n,o] = sum_k h(n,k) * Wm[o,k]
// ---------------------------------------------------------------------------
__global__ __launch_bounds__(256) void k_merge(const float* __restrict__ agg,
                                               const float* __restrict__ P,
                                               const float* __restrict__ dsrc,
                                               const float* __restrict__ bias,
                                               const float* __restrict__ Wm,
                                               float* __restrict__ out) {
    int wid = blockIdx.x * 8 + (threadIdx.x >> 5);
    if (wid >= (NN / 16) * 2) return;          // 3125 row-tiles * 2 col-tiles
    int tm   = wid >> 1;
    int tn   = wid & 1;
    int lane = threadIdx.x & 31;
    int half = lane >> 4;
    int lid  = lane & 15;

    int n  = tm * 16 + lid;                    // A row (node)
    int oc = tn * 16 + lid;                    // B col (final out channel)
    const float* wr = Wm + oc * (LL * FOUT);

    v8f acc = {};
    for (int k0 = 0; k0 < LL * FOUT; k0 += 4) {
        int ka = k0 + half * 2;                // even; ka,ka+1 share a layer
        int l  = ka >> 5;
        int f  = ka & 31;
        int vr = l * NN + n;
        int vp = (1 - l) * NN + n;             // cross-layer partner row
        float s  = rsqrtf(dsrc[vr]);
        float h0 = (agg[vr * FOUT + f]     + P[vr * FOUT + f]     + P[vp * FOUT + f])     * s
                   + bias[l * FOUT + f];
        float h1 = (agg[vr * FOUT + f + 1] + P[vr * FOUT + f + 1] + P[vp * FOUT + f + 1]) * s
                   + bias[l * FOUT + f + 1];
        h0 = h0 > 0.0f ? h0 : NEG * h0;
        h1 = h1 > 0.0f ? h1 : NEG * h1;
        v2f a; a.x = h0;     a.y = h1;
        v2f b; b.x = wr[ka]; b.y = wr[ka + 1];
        acc = __builtin_amdgcn_wmma_f32_16x16x4_f32(false, a, false, b,
                                                    (short)0, acc, false, false);
    }

#pragma unroll
    for (int v = 0; v < 8; ++v) {
        int node = tm * 16 + v + 8 * half;
        out[node * FOUT + oc] = acc[v];
    }
}

// ---------------------------------------------------------------------------
extern "C" void kernel_launch(void* const* d_in, const int* in_sizes, int n_in,
                              void* d_out, int out_size, void* d_ws, size_t ws_size,
                              hipStream_t stream) {
    (void)in_sizes; (void)n_in; (void)out_size; (void)ws_size;
    const float* x    = (const float*)d_in[0];
    const int*   e0   = (const int*)  d_in[1];
    const int*   e1   = (const int*)  d_in[2];
    const float* Wp   = (const float*)d_in[3];
    const float* Wm   = (const float*)d_in[4];
    const float* bias = (const float*)d_in[5];
    float*       out  = (float*)d_out;

    float* ws       = (float*)d_ws;
    float* deg_src  = ws;                     // 2*NN floats
    float* deg_trg  = ws + 2 * NN;            // 2*NN floats
    float* P        = ws + 4 * NN;            // 2*NN*FOUT floats
    float* agg      = P + 2 * NN * FOUT;      // 2*NN*FOUT floats
    // total workspace: 6.6M floats = 26.4 MB

    // 1) init degrees (=2.0 for implicit self+cross edges) and zero agg
    k_init<<<(2 * NN * FOUT + 255) / 256, 256, 0, stream>>>(deg_src, agg);

    // 2) degree counts over explicit edges
    k_deg<<<(2 * ELEN + 255) / 256, 256, 0, stream>>>(e0, e1, deg_src, deg_trg);

    // 3) projection GEMM (WMMA f32) with out-degree scaling
    k_proj<<<((NN / 16) * 4 + 7) / 8, 256, 0, stream>>>(x, Wp, deg_trg, P);

    // 4) scatter aggregation over explicit edges
    long long eThreads = (long long)2 * ELEN * FOUT;
    k_edge<<<(int)((eThreads + 255) / 256), 256, 0, stream>>>(e0, e1, P, agg);

    // 5) merge GEMM (WMMA f32) with fused self/cross/scale/bias/leaky-ReLU
    k_merge<<<((NN / 16) * 2 + 7) / 8, 256, 0, stream>>>(agg, P, deg_src, bias, Wm, out);
}